// TuckER_KBLN_46119358825174
// MI455X (gfx1250) — compile-verified
//
#include <hip/hip_runtime.h>
#include <hip/hip_bf16.h>

// ---------------------------------------------------------------------------
// TuckER + KBLN fused pipeline for gfx1250 (MI455X).
//   B=64, NE=40000, NR=500, D1=D2=200, L=64
// GEMMs use V_WMMA_F32_16X16X4_F32 (full fp32 precision, CDNA5 matrix pipe).
// KBLN branch uses LDS-tiled v_exp_f32 loop. global_prefetch_b8 warms epilogue.
// ---------------------------------------------------------------------------

#define BB   64
#define NE   40000
#define D1   200
#define D2   200
#define LL   64
#define BN_EPS 1e-5f
#define LOG2E  1.4426950408889634f

typedef __attribute__((ext_vector_type(2))) float v2f;
typedef __attribute__((ext_vector_type(8))) float v8f;

__device__ __forceinline__ v8f wmma_f32_16x16x4(v2f a, v2f b, v8f c) {
    // (neg_a, A, neg_b, B, c_mod, C, reuse_a, reuse_b)
    return __builtin_amdgcn_wmma_f32_16x16x4_f32(false, a, false, b, (short)0, c,
                                                 false, false);
}

// ---------------------------------------------------------------------------
// Prologue: bn0 on gathered E rows; gather n_h - c, nf[r_idx]; -log2e/var_l.
// One block, 256 threads.
// ---------------------------------------------------------------------------
__global__ void prep_kernel(const int* __restrict__ e1_idx,
                            const int* __restrict__ r_idx,
                            const float* __restrict__ E,
                            const float* __restrict__ lits,
                            const float* __restrict__ c,
                            const float* __restrict__ var_l,
                            const float* __restrict__ nf,
                            const float* __restrict__ g0,
                            const float* __restrict__ be0,
                            float* __restrict__ x0,
                            float* __restrict__ nhc,
                            float* __restrict__ wnf,
                            float* __restrict__ niv) {
    const int t = threadIdx.x;
    if (t < D1) {
        float mu = 0.0f, m2 = 0.0f;
        for (int b = 0; b < BB; ++b) {
            float v = E[(size_t)e1_idx[b] * D1 + t];
            mu += v; m2 += v * v;
        }
        mu *= (1.0f / BB); m2 *= (1.0f / BB);
        float rs = rsqrtf(m2 - mu * mu + BN_EPS);
        float g = g0[t], be = be0[t];
        for (int b = 0; b < BB; ++b) {
            float v = E[(size_t)e1_idx[b] * D1 + t];
            x0[b * D1 + t] = g * (v - mu) * rs + be;
        }
    }
    for (int i = t; i < BB * LL; i += blockDim.x) {
        int b = i >> 6, l = i & 63;
        nhc[i] = lits[(size_t)e1_idx[b] * LL + l] - c[l];
        wnf[i] = nf[(size_t)r_idx[b] * LL + l];
    }
    if (t < LL) niv[t] = -LOG2E / var_l[t];
}

// ---------------------------------------------------------------------------
// u[b, k*200 + i] = R[r_idx[b], k] * x0[b, i]   (64 x 40000)
// ---------------------------------------------------------------------------
__global__ void outer_kernel(const int* __restrict__ r_idx,
                             const float* __restrict__ R,
                             const float* __restrict__ x0,
                             float* __restrict__ u) {
    const int b = blockIdx.y;
    const int m = blockIdx.x * blockDim.x + threadIdx.x;
    if (m >= NE) return;
    const int k = m / D1;
    const int i = m - k * D1;
    u[(size_t)b * NE + m] = R[(size_t)r_idx[b] * D2 + k] * x0[b * D1 + i];
}

// ---------------------------------------------------------------------------
// GEMM 1: xout(64x200) = u(64x40000) @ Wflat(40000x200)     [WMMA f32 K=4]
// One wave per 16x16 tile. grid = (13 n-tiles, 4 m-tiles), block = 32.
// OOB columns handled branch-free: clamp index + multiply by mask.
// ---------------------------------------------------------------------------
__global__ void gemm_uW_kernel(const float* __restrict__ u,
                               const float* __restrict__ W,
                               float* __restrict__ xout) {
    const int lane  = threadIdx.x;          // 0..31
    const int m0    = blockIdx.y * 16;
    const int n0    = blockIdx.x * 16;
    const int row   = lane & 15;            // M for A, N for B
    const int kh    = (lane >> 4) * 2;      // K sub-offset 0 or 2
    const int K     = NE;                   // 40000
    const int n     = n0 + row;
    const int n_s   = (n < D1) ? n : (D1 - 1);   // always in-bounds
    const float msk = (n < D1) ? 1.0f : 0.0f;    // branch-free zeroing
    v8f acc = {};
    const float* __restrict__ arow = u + (size_t)(m0 + row) * K;
    const float* __restrict__ bcol = W + n_s;    // column base, stride D1
#pragma unroll 4
    for (int k0 = 0; k0 < K; k0 += 4) {
        v2f a;
        a.x = arow[k0 + kh + 0];
        a.y = arow[k0 + kh + 1];
        v2f b;
        b.x = bcol[(size_t)(k0 + kh + 0) * D1] * msk;
        b.y = bcol[(size_t)(k0 + kh + 1) * D1] * msk;
        acc = wmma_f32_16x16x4(a, b, acc);
    }
    // C/D layout: VGPR v -> M = m0 + v + 8*(lane/16), N = n0 + lane%16
    if (n < D1) {
        const int hi = (lane >> 4) * 8;
#pragma unroll
        for (int v = 0; v < 8; ++v) {
            xout[(m0 + v + hi) * D1 + n] = acc[v];
        }
    }
}

// ---------------------------------------------------------------------------
// bn1 on xout(64x200) -> x1
// ---------------------------------------------------------------------------
__global__ void bn1_kernel(const float* __restrict__ xin,
                           const float* __restrict__ g1,
                           const float* __restrict__ be1,
                           float* __restrict__ x1) {
    const int t = threadIdx.x;
    if (t >= D1) return;
    float mu = 0.0f, m2 = 0.0f;
    for (int b = 0; b < BB; ++b) {
        float v = xin[b * D1 + t];
        mu += v; m2 += v * v;
    }
    mu *= (1.0f / BB); m2 *= (1.0f / BB);
    float rs = rsqrtf(m2 - mu * mu + BN_EPS);
    float g = g1[t], be = be1[t];
    for (int b = 0; b < BB; ++b) {
        float v = xin[b * D1 + t];
        x1[b * D1 + t] = g * (v - mu) * rs + be;
    }
}

// ---------------------------------------------------------------------------
// GEMM 2: score_l(64x40000) = x1(64x200) @ E^T   (B[k][n] = E[n][k])
// Each wave computes TWO adjacent 16-wide N-tiles sharing the A fragment.
// grid = (NE/32 n-pairs, 4 m-tiles), block = 32. K=200 = 50 wmma steps, exact.
// ---------------------------------------------------------------------------
__global__ void gemm_xE_kernel(const float* __restrict__ x1,
                               const float* __restrict__ E,
                               float* __restrict__ score_l) {
    const int lane = threadIdx.x;
    const int n0   = blockIdx.x * 32;
    const int m0   = blockIdx.y * 16;
    const int row  = lane & 15;
    const int kh   = (lane >> 4) * 2;
    v8f acc0 = {};
    v8f acc1 = {};
    const float* __restrict__ arow  = x1 + (size_t)(m0 + row) * D1;
    const float* __restrict__ brow0 = E  + (size_t)(n0 + row) * D1;       // E[n][*]
    const float* __restrict__ brow1 = E  + (size_t)(n0 + 16 + row) * D1;
#pragma unroll 4
    for (int k0 = 0; k0 < D1; k0 += 4) {
        v2f a;
        a.x = arow[k0 + kh + 0];
        a.y = arow[k0 + kh + 1];
        v2f b0;
        b0.x = brow0[k0 + kh + 0];
        b0.y = brow0[k0 + kh + 1];
        v2f b1;
        b1.x = brow1[k0 + kh + 0];
        b1.y = brow1[k0 + kh + 1];
        acc0 = wmma_f32_16x16x4(a, b0, acc0);
        acc1 = wmma_f32_16x16x4(a, b1, acc1);
    }
    const int hi = (lane >> 4) * 8;
    const int n  = n0 + row;
#pragma unroll
    for (int v = 0; v < 8; ++v) {
        score_l[(size_t)(m0 + v + hi) * NE + n]      = acc0[v];
        score_l[(size_t)(m0 + v + hi) * NE + n + 16] = acc1[v];
    }
}

// ---------------------------------------------------------------------------
// KBLN + sigmoid epilogue.
// Block handles 32 entities x all 64 batch rows. 256 threads: lane group bg
// (t>>5) covers b = bg*8 .. bg*8+7, nl = t&31 selects entity.
// score_n[b,n] = sum_l wnf[b,l] * exp2(niv[l] * (nhc[b,l] - lits[n,l])^2)
// ---------------------------------------------------------------------------
#define NTILE 32
__global__ void kbln_kernel(const float* __restrict__ lits,
                            const float* __restrict__ nhc,
                            const float* __restrict__ wnf,
                            const float* __restrict__ niv,
                            const float* __restrict__ score_l,
                            float* __restrict__ out) {
    __shared__ float s_lits[NTILE][LL + 1];   // +1 pad: kill stride-64 bank conflict
    __shared__ float s_nhc[BB][LL];           // broadcast reads within a wave
    __shared__ float s_wnf[BB][LL];
    __shared__ float s_niv[LL];

    const int t      = threadIdx.x;           // 0..255
    const int n_base = blockIdx.x * NTILE;
    const int nl     = t & (NTILE - 1);
    const int bg     = t >> 5;                // same for all lanes of a wave
    const int n      = n_base + nl;

    // warm L2 for the epilogue score_l reads while we grind the exp loop
    __builtin_prefetch(score_l + (size_t)(bg * 8) * NE + n, 0, 0);

    for (int i = t; i < NTILE * LL; i += 256) {
        int nn = i >> 6, l = i & 63;
        s_lits[nn][l] = lits[(size_t)(n_base + nn) * LL + l];
    }
    for (int i = t; i < BB * LL; i += 256) {
        int b = i >> 6, l = i & 63;
        s_nhc[b][l] = nhc[i];
        s_wnf[b][l] = wnf[i];
    }
    if (t < LL) s_niv[t] = niv[t];
    __syncthreads();

    float acc[8] = {0.f, 0.f, 0.f, 0.f, 0.f, 0.f, 0.f, 0.f};
    for (int l = 0; l < LL; ++l) {
        const float lit = s_lits[nl][l];
        const float nv  = s_niv[l];
#pragma unroll
        for (int j = 0; j < 8; ++j) {
            const int b = bg * 8 + j;
            float d = s_nhc[b][l] - lit;
            acc[j] += s_wnf[b][l] * __builtin_amdgcn_exp2f(nv * d * d);
        }
    }
#pragma unroll
    for (int j = 0; j < 8; ++j) {
        const int b = bg * 8 + j;
        float s = score_l[(size_t)b * NE + n] + acc[j];
        out[(size_t)b * NE + n] =
            1.0f / (1.0f + __builtin_amdgcn_exp2f(-LOG2E * s));
    }
}

// ---------------------------------------------------------------------------
// Launcher
// Inputs (setup_inputs order):
//  0 e1_idx(i32,64) 1 r_idx(i32,64) 2 E(f32,NE*200) 3 R(f32,500*200)
//  4 W(f32,200*200*200) 5 lits(f32,NE*64) 6 c(f32,64) 7 var_l(f32,64)
//  8 nf(f32,500*64) 9 bn0_gamma 10 bn0_beta 11 bn1_gamma 12 bn1_beta
// ---------------------------------------------------------------------------
extern "C" void kernel_launch(void* const* d_in, const int* in_sizes, int n_in,
                              void* d_out, int out_size, void* d_ws, size_t ws_size,
                              hipStream_t stream) {
    (void)in_sizes; (void)n_in; (void)out_size; (void)ws_size;

    const int*   e1_idx = (const int*)  d_in[0];
    const int*   r_idx  = (const int*)  d_in[1];
    const float* E      = (const float*)d_in[2];
    const float* R      = (const float*)d_in[3];
    const float* W      = (const float*)d_in[4];
    const float* lits   = (const float*)d_in[5];
    const float* c      = (const float*)d_in[6];
    const float* var_l  = (const float*)d_in[7];
    const float* nf     = (const float*)d_in[8];
    const float* g0     = (const float*)d_in[9];
    const float* be0    = (const float*)d_in[10];
    const float* g1     = (const float*)d_in[11];
    const float* be1    = (const float*)d_in[12];
    float*       out    = (float*)d_out;

    // workspace carve-up (floats)
    float* ws   = (float*)d_ws;
    float* x0   = ws;                         // 64*200   = 12800
    float* xo   = ws + 12800;                 // 64*200   = 12800
    float* x1   = ws + 25600;                 // 64*200   = 12800
    float* nhc  = ws + 38400;                 // 64*64    = 4096
    float* wnf  = ws + 42496;                 // 64*64    = 4096
    float* niv  = ws + 46592;                 // 64
    float* u    = ws + 46656;                 // 64*40000 = 2,560,000
    float* sl   = u  + (size_t)BB * NE;       // 64*40000 = 2,560,000
    // total ~20.7 MB

    // 1. bn0 + gathers
    prep_kernel<<<1, 256, 0, stream>>>(e1_idx, r_idx, E, lits, c, var_l, nf,
                                       g0, be0, x0, nhc, wnf, niv);

    // 2. outer product u = r ⊗ x0
    outer_kernel<<<dim3((NE + 255) / 256, BB), 256, 0, stream>>>(r_idx, R, x0, u);

    // 3. xout = u @ Wflat     (WMMA f32)
    gemm_uW_kernel<<<dim3(13, 4), 32, 0, stream>>>(u, W, xo);

    // 4. bn1
    bn1_kernel<<<1, 256, 0, stream>>>(xo, g1, be1, x1);

    // 5. score_l = x1 @ E^T   (WMMA f32, 2 N-tiles per wave)
    gemm_xE_kernel<<<dim3(NE / 32, 4), 32, 0, stream>>>(x1, E, sl);

    // 6. KBLN + sigmoid
    kbln_kernel<<<NE / NTILE, 256, 0, stream>>>(lits, nhc, wnf, niv, sl, out);
}